// DMTetGeometryFixedTopo_36833639530811
// MI455X (gfx1250) — compile-verified
//
#include <hip/hip_runtime.h>
#include <math.h>

#define TPB 256
#define EMPTY_KEY 0xffffffffffffffffull

// ---------------------------------------------------------------- CDNA5 TDM
#if defined(__HIP_DEVICE_COMPILE__) && defined(__gfx1250__) && \
    __has_builtin(__builtin_amdgcn_tensor_load_to_lds) &&       \
    __has_builtin(__builtin_amdgcn_s_wait_tensorcnt)
#define HAVE_TDM 1
#else
#define HAVE_TDM 0
#endif

#if HAVE_TDM
typedef __attribute__((ext_vector_type(4))) unsigned int tdm_g0_t;
typedef __attribute__((ext_vector_type(8))) int          tdm_g1_t;
typedef __attribute__((ext_vector_type(4))) int          tdm_g2_t;

// 1-D tile of 4-byte elements, global -> LDS, then wait TENSORcnt==0.
// Descriptor layout per CDNA5 ISA.
// 6-arg builtin form (amdgpu-toolchain clang-23 / therock-10.0 headers).
__device__ __forceinline__ void tdm_load_1d(const void* gsrc, void* lds_dst,
                                            unsigned tile_elems,
                                            unsigned tensor_elems) {
  unsigned long long ga = (unsigned long long)(size_t)gsrc;
  unsigned lo = (unsigned)(size_t)lds_dst;      // LDS byte address
  tdm_g0_t g0;
  g0[0] = 1u;                                   // count=1 (valid), user mode
  g0[1] = lo;                                   // lds_addr
  g0[2] = (unsigned)(ga & 0xffffffffu);         // global_addr[31:0]
  g0[3] = (unsigned)((ga >> 32) & 0x1ffffffu)   // global_addr[56:32]
        | (2u << 30);                           // type = 2 ("image")
  tdm_g1_t g1;
  g1[0] = (2 << 16);                            // data_size=4B, no multicast
  g1[1] = (int)((tensor_elems & 0xffffu) << 16);        // tensor_dim0 lo16
  g1[2] = (int)((tensor_elems >> 16) | (1u << 16));     // dim0 hi16 | dim1=1
  g1[3] = (int)((tile_elems & 0xffffu) << 16);          // tile_dim0
  g1[4] = 0;                                    // tile_dim1/2 unused (1-D)
  g1[5] = (int)tensor_elems;                    // tensor_dim0_stride lo32
  g1[6] = 0;
  g1[7] = 0;
  tdm_g2_t z4 = {0, 0, 0, 0};                   // groups 2/3 unused (<=2D)
  tdm_g1_t z8 = {0, 0, 0, 0, 0, 0, 0, 0};
  __builtin_amdgcn_tensor_load_to_lds(g0, g1, z4, z4, z8, 0);
  __builtin_amdgcn_s_wait_tensorcnt(0);
}
#endif

// ---------------------------------------------------------------- tables
__device__ __constant__ int TRI_TABLE_D[16][6] = {
  {-1,-1,-1,-1,-1,-1},{ 1, 0, 2,-1,-1,-1},{ 4, 0, 3,-1,-1,-1},{ 1, 4, 2, 1, 3, 4},
  { 3, 1, 5,-1,-1,-1},{ 2, 3, 0, 2, 5, 3},{ 1, 4, 0, 1, 5, 4},{ 4, 2, 5,-1,-1,-1},
  { 4, 5, 2,-1,-1,-1},{ 4, 1, 0, 4, 5, 1},{ 3, 2, 0, 3, 5, 2},{ 1, 3, 5,-1,-1,-1},
  { 4, 1, 2, 4, 3, 1},{ 3, 0, 4,-1,-1,-1},{ 2, 0, 1,-1,-1,-1},{-1,-1,-1,-1,-1,-1}};
__device__ __constant__ int NUM_TRI_D[16] = {0,1,1,2,1,2,2,1,1,2,2,1,2,1,1,0};
__device__ __constant__ int EDGE_A_D[6] = {0,0,0,1,1,2};
__device__ __constant__ int EDGE_B_D[6] = {1,2,3,2,3,3};

__device__ __forceinline__ unsigned long long mix64(unsigned long long x) {
  x += 0x9e3779b97f4a7c15ull;
  x = (x ^ (x >> 30)) * 0xbf58476d1ce4e5b9ull;
  x = (x ^ (x >> 27)) * 0x94d049bb133111ebull;
  return x ^ (x >> 31);
}

struct F3 { float x, y, z; };   // 12-byte packed triple -> b96 VMEM ops

// ---------------------------------------------------------------- kernels
// Pass A (defined first so the disasm snippet shows the TDM issue):
// crossing-edge dedup (hash CAS) fused with vertex interpolation.
__global__ void k_edges(const int* __restrict__ indices, int n_tets,
                        const float* __restrict__ sdf,
                        const float* __restrict__ vdef,
                        unsigned long long* __restrict__ keys,
                        int* __restrict__ vals, unsigned capMask,
                        unsigned* __restrict__ counters,
                        float* __restrict__ out_verts) {
  __shared__ int s_idx[TPB * 4];
  int base = blockIdx.x * TPB;
  unsigned remain = (unsigned)(n_tets - base) * 4u;
  unsigned tile = remain < (unsigned)(TPB * 4) ? remain : (unsigned)(TPB * 4);
#if HAVE_TDM
  if (threadIdx.x < 32u)                            // one wave drives the DMA
    tdm_load_1d(indices + (size_t)base * 4, s_idx, tile, remain);
#else
  for (unsigned i = threadIdx.x; i < tile; i += blockDim.x)
    s_idx[i] = indices[(size_t)base * 4 + i];
#endif
  if (threadIdx.x == 0)                             // stream-prefetch next slab
    __builtin_prefetch(indices + (size_t)(base + TPB) * 4, 0, 1);
  __syncthreads();
  __asm__ volatile("" ::: "memory");

  int tet = base + (int)threadIdx.x;
  if (tet >= n_tets) return;
  int vi[4]; float s[4]; int occ[4];
#pragma unroll
  for (int k = 0; k < 4; ++k) vi[k] = s_idx[threadIdx.x * 4 + k];
#pragma unroll
  for (int k = 0; k < 4; ++k) { s[k] = sdf[vi[k]]; occ[k] = s[k] > 0.0f; }
  int osum = occ[0] + occ[1] + occ[2] + occ[3];
  if (osum == 0 || osum == 4) return;               // tet not straddling

#pragma unroll
  for (int k = 0; k < 4; ++k) __builtin_prefetch(&vdef[(size_t)3 * vi[k]], 0, 3);

#pragma unroll
  for (int e = 0; e < 6; ++e) {
    int a = EDGE_A_D[e], b = EDGE_B_D[e];
    if (occ[a] == occ[b]) continue;                 // edge doesn't cross
    int va = vi[a], vb = vi[b];
    int lo = va < vb ? va : vb, hi = va < vb ? vb : va;
    unsigned long long key =
        ((unsigned long long)(unsigned)lo << 32) | (unsigned)hi;
    unsigned h = (unsigned)mix64(key) & capMask;
    for (;;) {
      unsigned long long prev = atomicCAS(&keys[h], EMPTY_KEY, key);
      if (prev == EMPTY_KEY) {                      // winner: assign + interp
        int id = (int)atomicAdd(&counters[0], 1u);
        vals[h] = id;
        float slo = sdf[lo], shi = sdf[hi];
        float inv = 1.0f / (slo - shi);
        float wlo = -shi * inv, whi = slo * inv;
        const F3* plo = (const F3*)&vdef[(size_t)3 * lo];
        const F3* phi = (const F3*)&vdef[(size_t)3 * hi];
        F3 pl = *plo, ph = *phi;
        F3 r;
        r.x = pl.x * wlo + ph.x * whi;
        r.y = pl.y * wlo + ph.y * whi;
        r.z = pl.z * wlo + ph.z * whi;
        *(F3*)&out_verts[(size_t)3 * id] = r;
        break;
      }
      if (prev == key) break;                       // duplicate edge
      h = (h + 1) & capMask;                        // linear probe
    }
  }
}

__device__ __forceinline__ int hash_lookup(
    const unsigned long long* __restrict__ keys, const int* __restrict__ vals,
    unsigned capMask, unsigned long long key) {
  unsigned h = (unsigned)mix64(key) & capMask;
  for (int p = 0; p < 65536; ++p) {
    unsigned long long k = keys[h];
    if (k == key) return vals[h];
    if (k == EMPTY_KEY) return -1;
    h = (h + 1) & capMask;
  }
  return -1;
}

// Pass B: marching-tet face emission into workspace (compacted via atomics).
__global__ void k_faces(const int* __restrict__ indices, int n_tets,
                        const float* __restrict__ sdf,
                        const unsigned long long* __restrict__ keys,
                        const int* __restrict__ vals, unsigned capMask,
                        unsigned* __restrict__ counters,
                        int* __restrict__ faceScratch,
                        int* __restrict__ fgScratch) {
  __shared__ int s_idx[TPB * 4];
  int base = blockIdx.x * TPB;
  unsigned remain = (unsigned)(n_tets - base) * 4u;
  unsigned tile = remain < (unsigned)(TPB * 4) ? remain : (unsigned)(TPB * 4);
#if HAVE_TDM
  if (threadIdx.x < 32u)
    tdm_load_1d(indices + (size_t)base * 4, s_idx, tile, remain);
#else
  for (unsigned i = threadIdx.x; i < tile; i += blockDim.x)
    s_idx[i] = indices[(size_t)base * 4 + i];
#endif
  __syncthreads();
  __asm__ volatile("" ::: "memory");

  int tet = base + (int)threadIdx.x;
  if (tet >= n_tets) return;
  int vi[4]; int occ[4];
#pragma unroll
  for (int k = 0; k < 4; ++k) vi[k] = s_idx[threadIdx.x * 4 + k];
#pragma unroll
  for (int k = 0; k < 4; ++k) occ[k] = sdf[vi[k]] > 0.0f;
  int tetindex = occ[0] + occ[1] * 2 + occ[2] * 4 + occ[3] * 8;
  int ntri = NUM_TRI_D[tetindex];
  if (ntri == 0) return;

  int eid[6];
#pragma unroll
  for (int e = 0; e < 6; ++e) {
    eid[e] = -1;
    int a = EDGE_A_D[e], b = EDGE_B_D[e];
    if (occ[a] == occ[b]) continue;
    int va = vi[a], vb = vi[b];
    int lo = va < vb ? va : vb, hi = va < vb ? vb : va;
    unsigned long long key =
        ((unsigned long long)(unsigned)lo << 32) | (unsigned)hi;
    eid[e] = hash_lookup(keys, vals, capMask, key);
  }
  unsigned f0 = atomicAdd(&counters[1], (unsigned)ntri);
  for (int t = 0; t < ntri; ++t) {
    size_t fb = (size_t)3 * (f0 + t);
    for (int k = 0; k < 3; ++k) {
      int slot = TRI_TABLE_D[tetindex][3 * t + k];
      faceScratch[fb + k] = (slot >= 0) ? eid[slot] : 0;
    }
    fgScratch[f0 + t] = tet * 2 + t;
  }
}

__global__ void k_init(unsigned long long* __restrict__ keys, unsigned cap,
                       unsigned* __restrict__ counters) {
  unsigned i = blockIdx.x * blockDim.x + threadIdx.x;
  if (i < 8u) counters[i] = 0u;
  if (i < cap) keys[i] = EMPTY_KEY;
}

// Fused pointwise: sdf = sign*|abs|, v_def = v + (2/(GRID_RES*2))*deform.
// b96 vector path for the xyz triples.
__global__ void k_pointwise(const float* __restrict__ verts,
                            const float* __restrict__ deform,
                            const float* __restrict__ ss,
                            const float* __restrict__ sa,
                            float* __restrict__ sdf, float* __restrict__ vdef,
                            int n_verts) {
  int i = blockIdx.x * blockDim.x + threadIdx.x;
  if (i >= n_verts) return;
  sdf[i] = ss[i] * fabsf(sa[i]);
  const float c = 2.0f / (128.0f * 2.0f);   // GRID_RES=128, DEFORM_SCALE=1
  F3 v = ((const F3*)verts)[i];
  F3 d = ((const F3*)deform)[i];
  F3 r; r.x = v.x + c * d.x; r.y = v.y + c * d.y; r.z = v.z + c * d.z;
  ((F3*)vdef)[i] = r;
}

// Pass C: lay faces + uv_idx into d_out once counts are final.
__global__ void k_emit(const int* __restrict__ faceScratch,
                       const int* __restrict__ fgScratch,
                       const unsigned* __restrict__ counters,
                       float* __restrict__ out, int Ng) {
  unsigned i = blockIdx.x * blockDim.x + threadIdx.x;
  unsigned nV = counters[0], nF = counters[1];
  if (i >= nF) return;
  size_t fbase = (size_t)3 * nV;
  out[fbase + 3 * i + 0] = (float)faceScratch[3 * (size_t)i + 0];
  out[fbase + 3 * i + 1] = (float)faceScratch[3 * (size_t)i + 1];
  out[fbase + 3 * i + 2] = (float)faceScratch[3 * (size_t)i + 2];
  // uv_idx follows uvs (Ng*Ng*4 rows * 2 floats)
  size_t uvibase = fbase + (size_t)3 * nF + (size_t)Ng * Ng * 8;
  int g = fgScratch[i];
  int tet = g >> 1, tri = g & 1;     // raw//Ng*Ng + raw%Ng == raw
  float* o = &out[uvibase + (size_t)3 * i];
  o[0] = (float)(tet * 4);
  o[1] = (float)(tet * 4 + tri + 1);
  o[2] = (float)(tet * 4 + tri + 2);
}

// Analytic UV atlas: 32 MB written once, never re-read -> non-temporal
// stores keep it from displacing L2-resident gather data.
__global__ void k_uvs(const unsigned* __restrict__ counters,
                      float* __restrict__ out, int Ng) {
  int j = blockIdx.x * blockDim.x + threadIdx.x;
  if (j >= Ng * Ng) return;
  unsigned nV = counters[0], nF = counters[1];
  size_t base = (size_t)3 * nV + (size_t)3 * nF + (size_t)j * 8;
  float inv = 1.0f / (float)Ng;
  float x = (float)(j % Ng) * inv;
  float y = (float)(j / Ng) * inv;
  float pad = 0.9f * inv;
  float* o = &out[base];
  __builtin_nontemporal_store(x,       o + 0);
  __builtin_nontemporal_store(y,       o + 1);
  __builtin_nontemporal_store(x + pad, o + 2);
  __builtin_nontemporal_store(y,       o + 3);
  __builtin_nontemporal_store(x + pad, o + 4);
  __builtin_nontemporal_store(y + pad, o + 5);
  __builtin_nontemporal_store(x,       o + 6);
  __builtin_nontemporal_store(y + pad, o + 7);
}

// ---------------------------------------------------------------- host
extern "C" void kernel_launch(void* const* d_in, const int* in_sizes, int n_in,
                              void* d_out, int out_size, void* d_ws,
                              size_t ws_size, hipStream_t stream) {
  (void)n_in; (void)out_size;
  const float* verts  = (const float*)d_in[0];
  const float* deform = (const float*)d_in[1];
  const float* ssign  = (const float*)d_in[2];
  const float* sabs   = (const float*)d_in[3];
  const int*   indices= (const int*)d_in[4];
  float* out = (float*)d_out;

  int n_verts = in_sizes[0] / 3;
  int n_tets  = in_sizes[4] / 4;
  int Ng = (int)ceil(sqrt((double)(((long long)n_tets * 2 + 1) / 2)));
  if (Ng < 1) Ng = 1;

  // workspace layout
  char* ws = (char*)d_ws;
  unsigned* counters = (unsigned*)ws;                         // 64 B
  float* sdf  = (float*)(ws + 64);                            // n_verts
  float* vdef = sdf + n_verts;                                // 3*n_verts
  int* faceScratch = (int*)(vdef + (size_t)3 * n_verts);      // 6*n_tets
  int* fgScratch   = faceScratch + (size_t)6 * n_tets;        // 2*n_tets
  char* heap = (char*)(fgScratch + (size_t)2 * n_tets);
  size_t used = (size_t)(heap - ws);
  used = (used + 7) & ~(size_t)7;
  size_t avail = ws_size > used ? ws_size - used : 0;
  unsigned cap = 1024u;
  while (((size_t)cap * 2u) * 12u <= avail && cap < (1u << 23)) cap <<= 1;
  unsigned long long* keys = (unsigned long long*)(ws + used);
  int* vals = (int*)(keys + cap);
  unsigned capMask = cap - 1u;

  int gInit  = (int)((cap + TPB - 1) / TPB);
  int gVert  = (n_verts + TPB - 1) / TPB;
  int gTet   = (n_tets + TPB - 1) / TPB;
  int gFace  = (2 * n_tets + TPB - 1) / TPB;
  int gUv    = (Ng * Ng + TPB - 1) / TPB;

  k_init     <<<gInit, TPB, 0, stream>>>(keys, cap, counters);
  k_pointwise<<<gVert, TPB, 0, stream>>>(verts, deform, ssign, sabs,
                                         sdf, vdef, n_verts);
  k_edges    <<<gTet,  TPB, 0, stream>>>(indices, n_tets, sdf, vdef,
                                         keys, vals, capMask, counters, out);
  k_faces    <<<gTet,  TPB, 0, stream>>>(indices, n_tets, sdf, keys, vals,
                                         capMask, counters, faceScratch,
                                         fgScratch);
  k_emit     <<<gFace, TPB, 0, stream>>>(faceScratch, fgScratch, counters,
                                         out, Ng);
  k_uvs      <<<gUv,   TPB, 0, stream>>>(counters, out, Ng);
}